// ExpertsFeedForward_76871324663968
// MI455X (gfx1250) — compile-verified
//
#include <hip/hip_runtime.h>

typedef float v2f __attribute__((ext_vector_type(2)));
typedef float v8f __attribute__((ext_vector_type(8)));

#define B_ 2
#define S_ 256
#define D_ 256
#define H_ 1024
#define E_ 16
#define K_ 2
#define NTOK (B_ * S_)            /* 512  */
#define NENT (NTOK * K_)          /* 1024 */
#define TILE_M 16
#define MAX_TILES (NENT / TILE_M) /* 64 */
#define XSTRIDE (D_ + 4)          /* 260: 16B-aligned rows, conflict-free banks */
#define GSTRIDE (H_ + 4)          /* 1028 */

__device__ __forceinline__ float gelu_tanh(float h) {
    float u = 0.7978845608028654f * (h + 0.044715f * h * h * h);
#if __has_builtin(__builtin_amdgcn_tanhf)
    float t = __builtin_amdgcn_tanhf(u);   // v_tanh_f32 (CDNA5 HW transcendental)
#else
    float t = tanhf(u);
#endif
    return 0.5f * h * (1.0f + t);
}

// ---------------- kernel 0: zero the output accumulator ----------------
__global__ void zero_out_kernel(float* __restrict__ out, int n) {
    int i = blockIdx.x * blockDim.x + threadIdx.x;
    if (i < n) out[i] = 0.0f;
}

// ---------------- kernel 1: deterministic expert bucketing ----------------
// One wave32 per expert scans all NENT entries in index order; ballot +
// prefix popcount gives a stable, deterministic ordering (no atomics).
__global__ __launch_bounds__(32) void build_buckets_kernel(
    const int* __restrict__ expert_indices,
    int* __restrict__ buckets, int* __restrict__ counts) {
    int e = blockIdx.x;
    int lane = threadIdx.x;
    int count = 0;
    for (int base = 0; base < NENT; base += 32) {
        int i = base + lane;
        bool m = (expert_indices[i] == e);
        unsigned mask = __builtin_amdgcn_ballot_w32(m);
        int pre = __builtin_popcount(mask & ((1u << lane) - 1u));
        if (m) buckets[e * NENT + count + pre] = i;
        count += __builtin_popcount(mask);
    }
    if (lane == 0) counts[e] = count;
}

// ---------------- kernel 2: per-(expert, 16-token tile) fused FFN ----------------
// 8 waves. Phase 1: hidden = X(16xD) * keys[e]^T -> 16xH via WMMA f32 16x16x4,
// GELU (v_tanh_f32) on accumulators, staged to LDS. Phase 2: out = g(16xH) *
// values[e]^T -> 16xD via WMMA, row-scaled, atomically accumulated.
__global__ __launch_bounds__(256) void moe_ffn_wmma_kernel(
    const float* __restrict__ x,
    const float* __restrict__ keys,
    const float* __restrict__ values,
    const float* __restrict__ expert_weights,
    const int* __restrict__ buckets,
    const int* __restrict__ counts,
    float* __restrict__ out) {
    __shared__ float xtile[TILE_M * XSTRIDE];
    __shared__ float gtile[TILE_M * GSTRIDE];
    __shared__ float wrow[TILE_M];
    __shared__ int trow[TILE_M];

    const int e = blockIdx.x / MAX_TILES;
    const int tile = blockIdx.x % MAX_TILES;
    const int cnt = counts[e];
    if (tile * TILE_M >= cnt) return;

    const int tid = threadIdx.x;
    const int wave = tid >> 5;
    const int lane = tid & 31;

    // tile metadata: token id + routing weight per row (pad rows -> weight 0)
    if (tid < TILE_M) {
        int j = tile * TILE_M + tid;
        if (j < cnt) {
            int ent = buckets[e * NENT + j];
            trow[tid] = ent / K_;
            wrow[tid] = expert_weights[ent];
        } else {
            trow[tid] = 0;
            wrow[tid] = 0.0f;
        }
    }
    __syncthreads();

    // stage X tile (16 x 256) into LDS with async global->LDS B128 copies
    // (ASYNCcnt-tracked; dsaddr = low 32 bits of the shared-memory address).
    for (int i = tid; i < TILE_M * (D_ / 4); i += 256) {
        int r = i >> 6;             // i / (D_/4)
        int c = (i & 63) << 2;      // 4-float chunk column
        unsigned lds_off = (unsigned)(size_t)(&xtile[r * XSTRIDE + c]);
        const float* gp = x + (size_t)trow[r] * D_ + c;
        asm volatile("global_load_async_to_lds_b128 %0, %1, off"
                     :: "v"(lds_off), "v"(gp) : "memory");
    }
    asm volatile("s_wait_asynccnt 0x0" ::: "memory");
    __syncthreads();

    // WMMA lane mapping (ISA 7.12.2):
    //  A 16x4 : row m = lane&15, k-base kb = (lane>>4)*2, 2 VGPRs {kb, kb+1}
    //  B 4x16 : col n = lane&15, rows {kb, kb+1}
    //  C 16x16: VGPR v -> row v + 8*(lane>>4), col lane&15
    const int m = lane & 15;
    const int n = lane & 15;
    const int kb = (lane >> 4) << 1;
    const int rbase = (lane >> 4) << 3;

    // -------- phase 1: hidden + GELU --------
    const float* ke = keys + (size_t)e * H_ * D_;
    for (int t = 0; t < 8; ++t) {
        const int h0 = (wave * 8 + t) * 16; // 8 waves x 8 tiles = H/16
        const float* brow = ke + (size_t)(h0 + n) * D_ + kb; // B[k][n] = keys[h0+n][k]
        __builtin_prefetch(brow, 0, 0); // global_prefetch_b8
        v8f acc = {};
        for (int d0 = 0; d0 < D_; d0 += 4) {
            v2f a = *(const v2f*)&xtile[m * XSTRIDE + d0 + kb];
            v2f b = *(const v2f*)(brow + d0);
            acc = __builtin_amdgcn_wmma_f32_16x16x4_f32(
                false, a, false, b, (short)0, acc, false, false);
        }
        for (int v = 0; v < 8; ++v) {
            gtile[(v + rbase) * GSTRIDE + h0 + n] = gelu_tanh(acc[v]);
        }
    }
    __syncthreads();

    // -------- phase 2: output projection --------
    const float* ve = values + (size_t)e * D_ * H_;
    for (int t = 0; t < 2; ++t) {
        const int d0 = (wave * 2 + t) * 16; // 8 waves x 2 tiles = D/16
        const float* brow = ve + (size_t)(d0 + n) * H_ + kb; // B[k][n] = values[d0+n][k]
        __builtin_prefetch(brow, 0, 0);
        v8f acc = {};
        for (int h0 = 0; h0 < H_; h0 += 4) {
            v2f a = *(const v2f*)&gtile[m * GSTRIDE + h0 + kb];
            v2f b = *(const v2f*)(brow + h0);
            acc = __builtin_amdgcn_wmma_f32_16x16x4_f32(
                false, a, false, b, (short)0, acc, false, false);
        }
        for (int v = 0; v < 8; ++v) {
            int r = v + rbase;
            atomicAdd(&out[(size_t)trow[r] * D_ + d0 + n], acc[v] * wrow[r]);
        }
    }
}

extern "C" void kernel_launch(void* const* d_in, const int* in_sizes, int n_in,
                              void* d_out, int out_size, void* d_ws, size_t ws_size,
                              hipStream_t stream) {
    const float* x = (const float*)d_in[0];
    const float* keys = (const float*)d_in[1];
    const float* values = (const float*)d_in[2];
    const float* expert_weights = (const float*)d_in[3];
    const int* expert_indices = (const int*)d_in[4];
    float* out = (float*)d_out;

    // workspace layout: [0,256B) counts, then buckets (E*NENT ints)
    int* counts = (int*)d_ws;
    int* buckets = (int*)((char*)d_ws + 256);

    zero_out_kernel<<<(out_size + 255) / 256, 256, 0, stream>>>(out, out_size);
    build_buckets_kernel<<<E_, 32, 0, stream>>>(expert_indices, buckets, counts);
    moe_ffn_wmma_kernel<<<E_ * MAX_TILES, 256, 0, stream>>>(
        x, keys, values, expert_weights, buckets, counts, out);
}